// ElementBlock1D_Lin_26010321944830
// MI455X (gfx1250) — compile-verified
//
#include <hip/hip_runtime.h>
#include <hip/hip_bf16.h>

// Bandwidth-bound FEM shape-function kernel for MI455X (gfx1250).
// ~0.6 GB total traffic, ~0.25 GFLOP -> AI ~0.4 FLOP/B, runtime floor ~27us
// at 23.3 TB/s. Strategy: 4 cells per lane so every output block is 16B
// aligned -> pure b128 non-temporal streaming stores; NT b128 index loads;
// regular-temporal coordinate gathers (2x reuse, fits L2); global_prefetch
// to run the index stream ahead.

namespace {

// Clang extended vector types (HIP_vector_type structs are rejected by the
// nontemporal builtins).
typedef __attribute__((ext_vector_type(4))) float v4f;
typedef __attribute__((ext_vector_type(4))) int   v4i;
typedef __attribute__((ext_vector_type(2))) int   v2i;

constexpr int BLOCK = 256;   // 8 waves per block (wave32)
constexpr int CPT   = 4;     // cells per thread -> 16B-aligned output blocks

__global__ __launch_bounds__(BLOCK) void fem1d_gauss3_kernel(
    const float* __restrict__ coords,   // (N_NODES)
    const int*   __restrict__ conn,     // (N_CELLS, 2) int32
    const int*   __restrict__ cell_id,  // (N_CELLS)    int32
    float*       __restrict__ outN,     // (N_CELLS, 2, 3)
    float*       __restrict__ outXg,    // (N_CELLS, 3)
    float*       __restrict__ outDJW,   // (N_CELLS, 3)
    int n_cells)
{
#pragma clang fp contract(off)
    const float GP[3] = { (float)(-0.7745966692414834), 0.0f,
                          (float)( 0.7745966692414834) };
    const float WG[3] = { (float)(5.0 / 9.0), (float)(8.0 / 9.0),
                          (float)(5.0 / 9.0) };

    const long long t    = (long long)blockIdx.x * BLOCK + threadIdx.x;
    const long long base = t * CPT;
    if (base >= n_cells) return;

    if (base + CPT <= n_cells) {
        // Keep the streaming index data ahead of the gathers (gfx1250
        // global_prefetch_b8; clamp to stay inside the allocation).
        long long pf = base + 16384;
        if (pf >= n_cells) pf = n_cells - 1;
        __builtin_prefetch(cell_id + pf, 0, 1);

        // One NT b128 load for 4 cell ids (read once, don't cache).
        const v4i cid = __builtin_nontemporal_load(
            reinterpret_cast<const v4i*>(cell_id + base));

        alignas(16) float nbuf [CPT * 6];
        alignas(16) float xgbuf[CPT * 3];
        alignas(16) float djbuf[CPT * 3];

#pragma unroll
        for (int k = 0; k < CPT; ++k) {
            const long long c = (long long)cid[k];
            // conn row c: byte offset 8*c -> always 8B aligned -> b64 load.
            const v2i cc = *reinterpret_cast<const v2i*>(conn + 2 * c);
            const float x1 = coords[cc.x - 1];   // temporal: 2x reuse, fits L2
            const float x2 = coords[cc.y - 1];
            const float d    = x2 - x1;
            const float detj = d * 0.5f;
#pragma unroll
            for (int j = 0; j < 3; ++j) {
                // Exact op order of the reference (contraction disabled).
                const float h  = ((GP[j] + 1.0f) * d) * 0.5f;
                const float xg = x1 + h;
                const float rc = (2.0f * (xg - x1)) / d - 1.0f;
                nbuf [k * 6 + j]     = -0.5f * rc + 0.5f;
                nbuf [k * 6 + 3 + j] =  0.5f * rc + 0.5f;
                xgbuf[k * 3 + j]     = xg;
                djbuf[k * 3 + j]     = detj * WG[j];
            }
        }

        // Outputs per lane: N = 96B @ byte offset 96t, x_g/detJW = 48B @ 48t:
        // all 16B aligned -> pure b128 non-temporal streaming stores.
        v4f* pN  = reinterpret_cast<v4f*>(outN   + base * 6);
        v4f* pXg = reinterpret_cast<v4f*>(outXg  + base * 3);
        v4f* pDj = reinterpret_cast<v4f*>(outDJW + base * 3);
        const v4f* sN  = reinterpret_cast<const v4f*>(nbuf);
        const v4f* sXg = reinterpret_cast<const v4f*>(xgbuf);
        const v4f* sDj = reinterpret_cast<const v4f*>(djbuf);
#pragma unroll
        for (int q = 0; q < 6; ++q) __builtin_nontemporal_store(sN[q],  pN  + q);
#pragma unroll
        for (int q = 0; q < 3; ++q) __builtin_nontemporal_store(sXg[q], pXg + q);
#pragma unroll
        for (int q = 0; q < 3; ++q) __builtin_nontemporal_store(sDj[q], pDj + q);
    } else {
        // Scalar tail (only if n_cells % 4 != 0).
        for (long long p = base; p < n_cells; ++p) {
            const long long c = (long long)cell_id[p];
            const float x1 = coords[conn[2 * c]     - 1];
            const float x2 = coords[conn[2 * c + 1] - 1];
            const float d    = x2 - x1;
            const float detj = d * 0.5f;
            for (int j = 0; j < 3; ++j) {
                const float h  = ((GP[j] + 1.0f) * d) * 0.5f;
                const float xg = x1 + h;
                const float rc = (2.0f * (xg - x1)) / d - 1.0f;
                outN  [p * 6 + j]     = -0.5f * rc + 0.5f;
                outN  [p * 6 + 3 + j] =  0.5f * rc + 0.5f;
                outXg [p * 3 + j]     = xg;
                outDJW[p * 3 + j]     = detj * WG[j];
            }
        }
    }
}

} // namespace

extern "C" void kernel_launch(void* const* d_in, const int* in_sizes, int n_in,
                              void* d_out, int out_size, void* d_ws, size_t ws_size,
                              hipStream_t stream) {
    // setup_inputs order: 0=x, 1=coordinates, 2=nodal_values, 3=connectivity,
    // 4=cell_id. x / nodal_values are dead inputs in the reference.
    const float* coords  = (const float*)d_in[1];
    const int*   conn    = (const int*)d_in[3];   // JAX demotes int64 -> int32
    const int*   cell_id = (const int*)d_in[4];
    const int    n       = in_sizes[4];

    // Output tuple concatenated flat: N (6n) | x_g (3n) | detJ*WG (3n).
    float* outN   = (float*)d_out;
    float* outXg  = outN + (size_t)6 * n;
    float* outDJW = outN + (size_t)9 * n;

    const long long threads = ((long long)n + CPT - 1) / CPT;
    const int blocks = (int)((threads + BLOCK - 1) / BLOCK);

    fem1d_gauss3_kernel<<<blocks, BLOCK, 0, stream>>>(
        coords, conn, cell_id, outN, outXg, outDJW, n);
}